// ODERNN_22660247454195
// MI455X (gfx1250) — compile-verified
//
#include <hip/hip_runtime.h>

// ---------------- problem constants ----------------
namespace {
constexpr int B_ = 512, K_ = 1024, U_ = 16, P_ = 32, H_ = 128, L_ = 32, TD_ = 64;
constexpr int MT = 16;          // batch rows per block
constexpr int NTHREADS = 128;   // 4 wave32

// LDS layout (float offsets). Total ~316KB < 320KB/WGP (CDNA5 only).
constexpr int OFF_WHH  = 0;                         // w_hh   [384][128]
constexpr int OFF_WIH  = OFF_WHH  + 3*H_*H_;        // w_ih   [384][32]
constexpr int OFF_HW   = OFF_WIH  + 3*H_*L_;        // head_W [64][128]
constexpr int OFF_LW   = OFF_HW   + TD_*H_;         // lift_W [32][48]
constexpr int OFF_JT   = OFF_LW   + L_*(U_+P_);     // jump^T [32][16]
constexpr int OFF_BIH  = OFF_JT   + P_*U_;          // b_ih [384]
constexpr int OFF_BHH  = OFF_BIH  + 3*H_;           // b_hh [384]
constexpr int OFF_HB   = OFF_BHH  + 3*H_;           // head_b [64]
constexpr int OFF_LB   = OFF_HB   + TD_;            // lift_b [32]
constexpr int OFF_H0   = OFF_LB   + L_;             // h buf 0 [16][128]
constexpr int OFF_H1   = OFF_H0   + MT*H_;          // h buf 1 [16][128]
constexpr int OFF_FEAT = OFF_H1   + MT*H_;          // feat [16][48]
constexpr int OFF_X    = OFF_FEAT + MT*(U_+P_);     // x    [16][32]
constexpr int OFF_TH   = OFF_X    + MT*L_;          // theta[16][64]
constexpr int OFF_DT   = OFF_TH   + MT*TD_;         // dt   [16]
constexpr int LDS_FLOATS = OFF_DT + MT;             // 78960 floats = 315840 B
}

typedef float v2f __attribute__((ext_vector_type(2)));
typedef float v8f __attribute__((ext_vector_type(8)));

// Full-precision fp32 WMMA: D(16x16) = A(16x4) * B(4x16) + C
__device__ __forceinline__ v8f wmma4(v2f a, v2f b, v8f c) {
  return __builtin_amdgcn_wmma_f32_16x16x4_f32(
      /*neg_a=*/false, a, /*neg_b=*/false, b,
      /*c_mod=*/(short)0, c, /*reuse_a=*/false, /*reuse_b=*/false);
}
__device__ __forceinline__ float sigmoidf_(float x) { return 1.0f / (1.0f + __expf(-x)); }

extern "C" __global__ __launch_bounds__(NTHREADS)
void odernn_scan_kernel(const float* __restrict__ y0,
                        const float* __restrict__ u_seq,
                        const float* __restrict__ dt_seq,
                        const float* __restrict__ y_seq,
                        const float* __restrict__ jump,
                        const float* __restrict__ lift_W,
                        const float* __restrict__ lift_b,
                        const float* __restrict__ w_ih,
                        const float* __restrict__ w_hh,
                        const float* __restrict__ b_ih,
                        const float* __restrict__ b_hh,
                        const float* __restrict__ head_W,
                        const float* __restrict__ head_b,
                        const int*   __restrict__ tf_every_p,
                        float* __restrict__ ys_out,
                        float* __restrict__ th_out)
{
  extern __shared__ float s[];
  const int tid = threadIdx.x;

  // ---- preload all weights into LDS (resident for the whole 1024-step scan) ----
  for (int i = tid; i < 3*H_*H_;    i += NTHREADS) s[OFF_WHH+i] = w_hh[i];
  for (int i = tid; i < 3*H_*L_;    i += NTHREADS) s[OFF_WIH+i] = w_ih[i];
  for (int i = tid; i < TD_*H_;     i += NTHREADS) s[OFF_HW+i]  = head_W[i];
  for (int i = tid; i < L_*(U_+P_); i += NTHREADS) s[OFF_LW+i]  = lift_W[i];
  for (int i = tid; i < U_*P_;      i += NTHREADS) {      // transpose jump -> [P][U]
    int kk = i >> 5, nn = i & 31;
    s[OFF_JT + nn*U_ + kk] = jump[i];
  }
  for (int i = tid; i < 3*H_; i += NTHREADS) { s[OFF_BIH+i] = b_ih[i]; s[OFF_BHH+i] = b_hh[i]; }
  if (tid < TD_) s[OFF_HB+tid] = head_b[tid];
  if (tid < L_)  s[OFF_LB+tid] = lift_b[tid];
  for (int i = tid; i < MT*H_; i += NTHREADS) s[OFF_H0+i] = 0.0f;   // h(0) = 0
  const int tfe = tf_every_p[0];

  const int wv   = tid >> 5;        // wave id 0..3 (uniform per wave)
  const int lane = tid & 31;
  const int lm   = lane & 15;
  const int half = lane >> 4;
  const int d2   = half * 2;        // fp32 WMMA A/B K-offset per lane half
  const int rowbase = (int)blockIdx.x * MT;

  v8f yfr, hfr0, hfr1;
  for (int r = 0; r < 8; ++r) { yfr[r] = 0.f; hfr0[r] = 0.f; hfr1[r] = 0.f; }
  if (wv < 2) {   // waves 0/1 carry y fragments (cols 16*wv .. 16*wv+15)
    for (int r = 0; r < 8; ++r)
      yfr[r] = y0[(size_t)(rowbase + r + 8*half)*P_ + wv*16 + lm];
  }
  __syncthreads();

  for (int k = 0; k < K_; ++k) {
    const int curo = (k & 1) ? OFF_H1 : OFF_H0;   // h_prev buffer
    const int nxto = (k & 1) ? OFF_H0 : OFF_H1;   // h_new buffer

    // ---- stage 1: gather per-step inputs into LDS ----
    {
      int idx = tid * 2;                      // 256 u values / 128 threads
      int m = idx >> 4, c = idx & 15;
      const float* up = &u_seq[((size_t)(rowbase+m)*K_ + k)*U_ + c];
      v2f uv = *(const v2f*)up;
      s[OFF_FEAT + m*48 + c]     = uv.x;
      s[OFF_FEAT + m*48 + c + 1] = uv.y;
      if (k + 1 < K_) __builtin_prefetch(up + U_, 0, 0);   // global_prefetch next step
    }
    if (tid < MT) s[OFF_DT + tid] = dt_seq[(size_t)(rowbase+tid)*K_ + k];
    const bool tf = (k > 0) && ((k % tfe) == 0);
    if (tf) {
      for (int i = tid; i < MT*P_; i += NTHREADS) {
        int m = i >> 5, p = i & 31;
        s[OFF_FEAT + m*48 + U_ + p] = y_seq[((size_t)(rowbase+m)*K_ + (k-1))*P_ + p];
      }
    } else if (wv < 2) {
      for (int r = 0; r < 8; ++r)
        s[OFF_FEAT + (r+8*half)*48 + U_ + wv*16 + lm] = yfr[r];
    }
    __syncthreads();

    // ---- stage 2: x = silu(feat @ lift_W.T + lift_b)  (waves 0-1, one 16-col tile each) ----
    if (wv < 2) {
      const int n = wv*16 + lm;
      const float bias = s[OFF_LB + n];
      v8f c; for (int r = 0; r < 8; ++r) c[r] = bias;
      for (int k0 = 0; k0 < U_+P_; k0 += 4) {
        v2f a = *(const v2f*)&s[OFF_FEAT + lm*48 + k0 + d2];
        v2f b = *(const v2f*)&s[OFF_LW   + n*48  + k0 + d2];
        c = wmma4(a, b, c);
      }
      for (int r = 0; r < 8; ++r) {
        float v = c[r];
        s[OFF_X + (r+8*half)*L_ + n] = v * sigmoidf_(v);
      }
    }
    __syncthreads();

    // ---- stage 3: GRU gates; each wave owns H-columns [32*wv, 32*wv+32) ----
    {
      v8f accR[2], accZ[2], accNi[2], accNh[2];
      for (int t = 0; t < 2; ++t) {
        const int n = 32*wv + 16*t + lm;
        float br  = s[OFF_BIH + n]        + s[OFF_BHH + n];
        float bz  = s[OFF_BIH + H_ + n]   + s[OFF_BHH + H_ + n];
        float bni = s[OFF_BIH + 2*H_ + n];
        float bnh = s[OFF_BHH + 2*H_ + n];
        for (int r = 0; r < 8; ++r) { accR[t][r]=br; accZ[t][r]=bz; accNi[t][r]=bni; accNh[t][r]=bnh; }
      }
      // gi = x @ w_ih.T   (K = 32)
      for (int k0 = 0; k0 < L_; k0 += 4) {
        v2f a = *(const v2f*)&s[OFF_X + lm*L_ + k0 + d2];
        for (int t = 0; t < 2; ++t) {
          const int n = 32*wv + 16*t + lm;
          v2f b0 = *(const v2f*)&s[OFF_WIH + (n)*L_        + k0 + d2];
          v2f b1 = *(const v2f*)&s[OFF_WIH + (H_+n)*L_     + k0 + d2];
          v2f b2 = *(const v2f*)&s[OFF_WIH + (2*H_+n)*L_   + k0 + d2];
          accR[t]  = wmma4(a, b0, accR[t]);
          accZ[t]  = wmma4(a, b1, accZ[t]);
          accNi[t] = wmma4(a, b2, accNi[t]);
        }
      }
      // gh = h_prev @ w_hh.T   (K = 128, the dominant GEMM)
      for (int k0 = 0; k0 < H_; k0 += 4) {
        v2f a = *(const v2f*)&s[curo + lm*H_ + k0 + d2];
        for (int t = 0; t < 2; ++t) {
          const int n = 32*wv + 16*t + lm;
          v2f b0 = *(const v2f*)&s[OFF_WHH + (n)*H_        + k0 + d2];
          v2f b1 = *(const v2f*)&s[OFF_WHH + (H_+n)*H_     + k0 + d2];
          v2f b2 = *(const v2f*)&s[OFF_WHH + (2*H_+n)*H_   + k0 + d2];
          accR[t]  = wmma4(a, b0, accR[t]);
          accZ[t]  = wmma4(a, b1, accZ[t]);
          accNh[t] = wmma4(a, b2, accNh[t]);   // h_n kept separate (n = tanh(i_n + r*h_n))
        }
      }
      for (int t = 0; t < 2; ++t) {
        v8f hprev = t ? hfr1 : hfr0;
        v8f hnew;
        for (int r = 0; r < 8; ++r) {
          float rg = sigmoidf_(accR[t][r]);
          float zg = sigmoidf_(accZ[t][r]);
          float ng = tanhf(accNi[t][r] + rg * accNh[t][r]);
          hnew[r] = (1.0f - zg) * ng + zg * hprev[r];
        }
        const int n = 32*wv + 16*t + lm;
        for (int r = 0; r < 8; ++r) s[nxto + (r+8*half)*H_ + n] = hnew[r];
        if (t) hfr1 = hnew; else hfr0 = hnew;
      }
    }
    __syncthreads();

    // ---- stage 4: theta = lo + (hi-lo)*sigmoid(h_new @ head_W.T + head_b) ----
    {
      const int n = 16*wv + lm;                 // 4 waves x 16 cols = TD
      const float hb = s[OFF_HB + n];
      v8f c; for (int r = 0; r < 8; ++r) c[r] = hb;
      for (int k0 = 0; k0 < H_; k0 += 4) {
        v2f a = *(const v2f*)&s[nxto   + lm*H_ + k0 + d2];
        v2f b = *(const v2f*)&s[OFF_HW + n*H_  + k0 + d2];
        c = wmma4(a, b, c);
      }
      for (int r = 0; r < 8; ++r) {
        int m = r + 8*half;
        float th = 0.001f + (2.0f - 0.001f) * sigmoidf_(c[r]);
        s[OFF_TH + m*TD_ + n] = th;
        th_out[((size_t)(rowbase+m)*K_ + k)*TD_ + n] = th;
      }
    }
    __syncthreads();

    // ---- stage 5: y = RK4(y_prev + u_k @ jump, dt, theta)  (waves 0-1) ----
    if (wv < 2) {
      const int n = 16*wv + lm;
      v8f c = yfr;                               // y_prev as the C operand
      for (int k0 = 0; k0 < U_; k0 += 4) {
        v2f a = *(const v2f*)&s[OFF_FEAT + lm*48 + k0 + d2];   // u_k
        v2f b = *(const v2f*)&s[OFF_JT   + n*U_  + k0 + d2];
        c = wmma4(a, b, c);
      }
      for (int r = 0; r < 8; ++r) {
        int m = r + 8*half;
        float av = s[OFF_TH + m*TD_ + n];        // a = theta[:, :P]
        float bv = s[OFF_TH + m*TD_ + P_ + n];   // b = theta[:, P:]
        float dt = s[OFF_DT + m];
        float y  = c[r];
        float k1 = bv - av*y;
        float y2 = y + 0.5f*dt*k1; float k2 = bv - av*y2;
        float y3 = y + 0.5f*dt*k2; float k3 = bv - av*y3;
        float y4 = y + dt*k3;      float k4 = bv - av*y4;
        y = y + dt*(1.0f/6.0f)*(k1 + 2.0f*k2 + 2.0f*k3 + k4);
        c[r] = y;
        ys_out[((size_t)(rowbase+m)*K_ + k)*P_ + n] = y;
      }
      yfr = c;
    }
    __syncthreads();
  }
}

extern "C" void kernel_launch(void* const* d_in, const int* in_sizes, int n_in,
                              void* d_out, int out_size, void* d_ws, size_t ws_size,
                              hipStream_t stream) {
  (void)in_sizes; (void)n_in; (void)out_size; (void)d_ws; (void)ws_size;
  const float* y0      = (const float*)d_in[0];
  const float* u_seq   = (const float*)d_in[1];
  const float* dt_seq  = (const float*)d_in[2];
  const float* y_seq   = (const float*)d_in[3];
  const float* jump    = (const float*)d_in[4];
  const float* lift_W  = (const float*)d_in[5];
  const float* lift_b  = (const float*)d_in[6];
  const float* w_ih    = (const float*)d_in[7];
  const float* w_hh    = (const float*)d_in[8];
  const float* b_ih    = (const float*)d_in[9];
  const float* b_hh    = (const float*)d_in[10];
  const float* head_W  = (const float*)d_in[11];
  const float* head_b  = (const float*)d_in[12];
  const int*   tfe     = (const int*)d_in[13];

  float* out = (float*)d_out;
  float* ys  = out;                                  // [B,K,P]
  float* th  = out + (size_t)B_ * K_ * P_;           // [B,K,TD]

  const size_t shmem = (size_t)LDS_FLOATS * sizeof(float);   // ~316 KB, CDNA5 WGP LDS
  hipFuncSetAttribute(reinterpret_cast<const void*>(odernn_scan_kernel),
                      hipFuncAttributeMaxDynamicSharedMemorySize, (int)shmem);

  odernn_scan_kernel<<<dim3(B_ / MT), dim3(NTHREADS), shmem, stream>>>(
      y0, u_seq, dt_seq, y_seq, jump, lift_W, lift_b,
      w_ih, w_hh, b_ih, b_hh, head_W, head_b, tfe, ys, th);
}